// Attention_24008867184960
// MI455X (gfx1250) — compile-verified
//
#include <hip/hip_runtime.h>
#include <hip/hip_bf16.h>
#include <math.h>

// ---------------- CDNA5 WMMA f32 16x16x4 helpers ----------------
typedef float v2f __attribute__((ext_vector_type(2)));
typedef float v4f __attribute__((ext_vector_type(4)));
typedef float v8f __attribute__((ext_vector_type(8)));

__device__ __forceinline__ v8f wmma_f32(v2f a, v2f b, v8f c) {
  // V_WMMA_F32_16X16X4_F32 : D = A(16x4) x B(4x16) + C(16x16), exact fp32
  return __builtin_amdgcn_wmma_f32_16x16x4_f32(
      /*neg_a=*/false, a, /*neg_b=*/false, b,
      /*c_mod=*/(short)0, c, /*reuse_a=*/false, /*reuse_b=*/false);
}

// A-frag (16x4, row-major source, leading dim ld):
//   VGPR v, lanes 0-15: M=lane, K=v ; lanes 16-31: M=lane-16, K=v+2
__device__ __forceinline__ v2f load_a(const float* A, int m0, int k0, int ld, int lane) {
  const int half = lane >> 4, m = lane & 15;
  const float* p = A + (m0 + m) * ld + k0 + 2 * half;
  v2f r; r.x = p[0]; r.y = p[1]; return r;
}
// B-frag (4x16, row-major K x N source, leading dim ld):
//   VGPR v, lanes 0-15: K=v, N=lane ; lanes 16-31: K=v+2, N=lane-16
__device__ __forceinline__ v2f load_b(const float* B, int k0, int n0, int ld, int lane) {
  const int half = lane >> 4, n = lane & 15;
  const float* p = B + (k0 + 2 * half) * ld + n0 + n;
  v2f r; r.x = p[0]; r.y = p[ld]; return r;
}

static constexpr int T = 512, Bb = 8, D = 512, H = 8, DH = 64;
static constexpr int RROWS = 1024;      // 1023 distinct rel distances, padded
static constexpr int QKV_LD = 3 * D;    // 1536

// ---------------- 1) sinusoidal relative-position table ----------------
__global__ void sin_table_kernel(float* __restrict__ Remb) {
  const int r = blockIdx.x;
  const int c = threadIdx.x;             // 0..255
  const float dist = (float)(r - (T - 1));
  const float freq = __expf(-9.210340371976184f * (float)c * (1.0f / 256.0f));
  const float ang = dist * freq;
  Remb[r * D + c]       = sinf(ang);
  Remb[r * D + 256 + c] = cosf(ang);
}

// ---------------- 2) fp32 WMMA GEMM, register-blocked ----------------
// block = 256 threads (8 waves); block tile 256(M) x 64(N); each wave owns a
// 32x64 tile = 2x4 accumulators, so per K-step: 8 WMMA vs 2 b64 + 8 b32 loads.
__global__ void gemm_f32_wmma(const float* __restrict__ A, const float* __restrict__ B,
                              const float* __restrict__ bias, float* __restrict__ C,
                              int M, int N, int K) {
  const int tid = threadIdx.x, lane = tid & 31, wave = tid >> 5;
  const int n0 = blockIdx.x * 64;
  const int m0 = blockIdx.y * 256 + wave * 32;

  v8f acc[2][4];
  for (int mi = 0; mi < 2; ++mi)
    for (int ni = 0; ni < 4; ++ni)
      acc[mi][ni] = (v8f){0.f, 0.f, 0.f, 0.f, 0.f, 0.f, 0.f, 0.f};

#pragma unroll 4
  for (int k0 = 0; k0 < K; k0 += 4) {
    v2f a[2], b[4];
    for (int mi = 0; mi < 2; ++mi) a[mi] = load_a(A, m0 + 16 * mi, k0, K, lane);
    for (int ni = 0; ni < 4; ++ni) b[ni] = load_b(B, k0, n0 + 16 * ni, N, lane);
    for (int mi = 0; mi < 2; ++mi)
      for (int ni = 0; ni < 4; ++ni)
        acc[mi][ni] = wmma_f32(a[mi], b[ni], acc[mi][ni]);
  }

  const int half = lane >> 4, ln = lane & 15;
  for (int mi = 0; mi < 2; ++mi)
    for (int ni = 0; ni < 4; ++ni)
      for (int v = 0; v < 8; ++v) {
        const int row = m0 + 16 * mi + v + 8 * half;
        const int col = n0 + 16 * ni + ln;
        float r = acc[mi][ni][v];
        if (bias) r += bias[col];
        C[(size_t)row * N + col] = r;
      }
}

// ---------------- 3) pvr[r][h] = pos_v[h,:] . R[r,h,:] ----------------
__global__ void pvr_kernel(const float* __restrict__ R, const float* __restrict__ pos_v,
                           float* __restrict__ pvr) {
  const int g = blockIdx.x * blockDim.x + threadIdx.x;   // 1024*8 threads
  const int r = g >> 3, h = g & 7;
  float s = 0.f;
  const float* rp = R + r * D + h * DH;
  const float* pv = pos_v + h * DH;
  for (int d = 0; d < DH; ++d) s += pv[d] * rp[d];
  pvr[r * H + h] = s;
}

// ---------------- 4) flash attention with relative-shift BD ----------------
// grid (T/32, B, H), block 256 (8 waves). Dynamic LDS (floats):
//  qp 32x68 | kt 64x68 | vt 64x64 | S 32x68 | BD 32x548 | m/l/sc[32 ea] | red[256]
static constexpr int QP_LD = 68, KT_LD = 68, S_LD = 68, BD_LD = 548;
static constexpr int LDS_FLOATS =
    32 * QP_LD + 64 * KT_LD + 64 * 64 + 32 * S_LD + 32 * BD_LD + 96 + 256;

__global__ void attn_kernel(const float* __restrict__ qkv, const float* __restrict__ R,
                            const float* __restrict__ pvr, const float* __restrict__ pos_u,
                            const float* __restrict__ tau_log, float* __restrict__ attn_out) {
  extern __shared__ float smem[];
  float* qp = smem;                    // 32 x 68 : q rows (later q + pos_u)
  float* kt = qp + 32 * QP_LD;         // 64 x 68
  float* vt = kt + 64 * KT_LD;         // 64 x 64
  float* Sb = vt + 64 * 64;            // 32 x 68
  float* BD = Sb + 32 * S_LD;          // 32 x 548 (cols: r - i0, r = i-j+511)
  float* mrow = BD + 32 * BD_LD;       // 32
  float* lrow = mrow + 32;             // 32
  float* srow = lrow + 32;             // 32
  float* red  = srow + 32;             // 256

  const int i0 = blockIdx.x * 32;
  const int b  = blockIdx.y;
  const int h  = blockIdx.z;
  const int tid = threadIdx.x, lane = tid & 31, wave = tid >> 5;
  const int half = lane >> 4, ln = lane & 15;
  const float scl = 0.125f * __expf(tau_log[0]);   // dh^-0.5 * exp(log(1/tau))

  // load raw q tile (vectorized b128; strides 68 floats = 17*16B keep alignment)
  for (int e = tid; e < 32 * DH / 4; e += 256) {
    const int r = e >> 4, d4 = (e & 15) * 4;
    *(v4f*)(qp + r * QP_LD + d4) =
        *(const v4f*)(qkv + ((i0 + r) * Bb + b) * QKV_LD + h * DH + d4);
  }
  if (tid < 32) { mrow[tid] = -3.0e38f; lrow[tid] = 0.f; }
  __syncthreads();

  // BD[m][rr] = q[i0+m,:] . R[i0+rr, h, :] + pvr[i0+rr, h]
  // GEMM 32x544 (K=64); 68 subtiles of 16x16, wave-uniform loop
  for (int st = wave; st < 68; st += 8) {
    const int m0 = (st & 1) * 16;
    const int n0 = (st >> 1) * 16;
    v8f acc = {0.f, 0.f, 0.f, 0.f, 0.f, 0.f, 0.f, 0.f};
    for (int k0 = 0; k0 < DH; k0 += 4) {
      v2f a = load_a(qp, m0, k0, QP_LD, lane);
      // B[k][n] = R[(i0+n)*D + h*DH + k] -> two consecutive floats per lane
      const float* pb = R + (i0 + n0 + ln) * D + h * DH + k0 + 2 * half;
      v2f bf; bf.x = pb[0]; bf.y = pb[1];
      acc = wmma_f32(a, bf, acc);
    }
    for (int v = 0; v < 8; ++v) {
      const int row = m0 + v + 8 * half;
      const int col = n0 + ln;
      BD[row * BD_LD + col] = acc[v] + pvr[(i0 + col) * H + h];
    }
  }
  __syncthreads();
  // q' = q + pos_u (content bias folds into AC)
  for (int e = tid; e < 32 * DH / 4; e += 256) {
    const int r = e >> 4, d4 = (e & 15) * 4;
    v4f u = *(const v4f*)(pos_u + h * DH + d4);
    v4f* q4 = (v4f*)(qp + r * QP_LD + d4);
    *q4 = *q4 + u;
  }
  __syncthreads();

  // per-wave persistent O accumulator: wave -> (msub = wave>>2, nsub = wave&3)
  const int m0s = (wave >> 2) * 16, n0s = (wave & 3) * 16;
  v8f o_acc = {0.f, 0.f, 0.f, 0.f, 0.f, 0.f, 0.f, 0.f};

  for (int j0 = 0; j0 < T; j0 += 64) {
    // cooperative K/V tile load (b128)
    for (int e = tid; e < 64 * DH / 4; e += 256) {
      const int r = e >> 4, d4 = (e & 15) * 4;
      const float* base = qkv + ((j0 + r) * Bb + b) * QKV_LD + h * DH + d4;
      *(v4f*)(kt + r * KT_LD + d4) = *(const v4f*)(base + D);       // k
      *(v4f*)(vt + r * 64 + d4)    = *(const v4f*)(base + 2 * D);   // v
    }
    __syncthreads();

    // S = q' @ kt^T (16x16 subtile per wave) + BD gather, scale, diag mask
    {
      v8f acc = {0.f, 0.f, 0.f, 0.f, 0.f, 0.f, 0.f, 0.f};
      for (int k0 = 0; k0 < DH; k0 += 4) {
        v2f a = load_a(qp, m0s, k0, QP_LD, lane);
        const float* pb = kt + (n0s + ln) * KT_LD + k0 + 2 * half;  // B[k][n]=kt[n][k]
        v2f bf; bf.x = pb[0]; bf.y = pb[1];
        acc = wmma_f32(a, bf, acc);
      }
      for (int v = 0; v < 8; ++v) {
        const int row = m0s + v + 8 * half;
        const int col = n0s + ln;
        float val = (acc[v] + BD[row * BD_LD + (row - col + (T - 1) - j0)]) * scl;
        if (i0 + row == j0 + col) val = -3.0e38f;   // diag mask
        Sb[row * S_LD + col] = val;
      }
    }
    __syncthreads();

    // online softmax, 8 threads per row
    const int srw = tid >> 3, sub = tid & 7;
    {
      float pm = -3.0e38f;
      for (int c = sub; c < 64; c += 8) pm = fmaxf(pm, Sb[srw * S_LD + c]);
      red[tid] = pm;
    }
    __syncthreads();
    if (tid < 32) {
      const float mo = mrow[tid];
      float mx = mo;
      for (int s = 0; s < 8; ++s) mx = fmaxf(mx, red[tid * 8 + s]);
      mrow[tid] = mx;
      srow[tid] = __expf(mo - mx);
    }
    __syncthreads();
    {
      const float mx = mrow[srw];
      float ps = 0.f;
      for (int c = sub; c < 64; c += 8) {
        const float p = __expf(Sb[srw * S_LD + c] - mx);
        Sb[srw * S_LD + c] = p;
        ps += p;
      }
      red[tid] = ps;
    }
    __syncthreads();
    if (tid < 32) {
      float s = 0.f;
      for (int k = 0; k < 8; ++k) s += red[tid * 8 + k];
      lrow[tid] = lrow[tid] * srow[tid] + s;
    }
    __syncthreads();

    // O = O*sca + P @ V
    {
      v8f accp = {0.f, 0.f, 0.f, 0.f, 0.f, 0.f, 0.f, 0.f};
      for (int k0 = 0; k0 < 64; k0 += 4) {
        v2f a = load_a(Sb, m0s, k0, S_LD, lane);
        v2f bf = load_b(vt, k0, n0s, 64, lane);
        accp = wmma_f32(a, bf, accp);
      }
      for (int v = 0; v < 8; ++v) {
        const int row = m0s + v + 8 * half;
        o_acc[v] = o_acc[v] * srow[row] + accp[v];
      }
    }
    __syncthreads();
  }

  // epilogue: normalize and write (t, b, h*DH + dh)
  for (int v = 0; v < 8; ++v) {
    const int row = m0s + v + 8 * half;
    const int col = n0s + ln;
    attn_out[((i0 + row) * Bb + b) * D + h * DH + col] = o_acc[v] / lrow[row];
  }
}

// ---------------- host launch ----------------
extern "C" void kernel_launch(void* const* d_in, const int* in_sizes, int n_in,
                              void* d_out, int out_size, void* d_ws, size_t ws_size,
                              hipStream_t stream) {
  const float* x      = (const float*)d_in[0];   // (T,B,D)
  const float* W_qkv  = (const float*)d_in[1];   // (D,3D)
  const float* W_pos  = (const float*)d_in[2];   // (D,D)
  const float* pos_u  = (const float*)d_in[3];   // (H,DH)
  const float* pos_v  = (const float*)d_in[4];   // (H,DH)
  const float* W_out  = (const float*)d_in[5];   // (D,D)
  const float* b_out  = (const float*)d_in[6];   // (D,)
  const float* tau    = (const float*)d_in[7];   // (1,)
  float* out = (float*)d_out;

  // workspace layout (floats), total ~37.8 MB
  float* ws   = (float*)d_ws;
  float* Remb = ws;                               // 1024*512
  float* R    = Remb + RROWS * D;                 // 1024*512
  float* qkvb = R + RROWS * D;                    // 4096*1536
  float* pvr  = qkvb + (T * Bb) * QKV_LD;         // 1024*8
  float* aout = pvr + RROWS * H;                  // 4096*512

  // 1) sinusoid table for all 1023 relative distances (padded to 1024 rows)
  sin_table_kernel<<<RROWS, 256, 0, stream>>>(Remb);
  // 2) R = Remb @ W_pos   (1024x512 @ 512x512)
  gemm_f32_wmma<<<dim3(D / 64, RROWS / 256), 256, 0, stream>>>(Remb, W_pos, nullptr, R,
                                                               RROWS, D, D);
  // 3) qkv = x @ W_qkv    (4096x1536, K=512)
  gemm_f32_wmma<<<dim3(QKV_LD / 64, (T * Bb) / 256), 256, 0, stream>>>(x, W_qkv, nullptr,
                                                                       qkvb, T * Bb, QKV_LD, D);
  // 4) pvr[r][h] = pos_v[h] . R[r,h,:]
  pvr_kernel<<<(RROWS * H) / 256, 256, 0, stream>>>(R, pos_v, pvr);
  // 5) flash attention with relative-shift BD term
  const size_t lds_bytes = (size_t)LDS_FLOATS * sizeof(float);   // ~120 KB (<320 KB/WGP)
  attn_kernel<<<dim3(T / 32, Bb, H), 256, lds_bytes, stream>>>(qkvb, R, pvr, pos_u, tau, aout);
  // 6) out = aout @ W_out + b_out
  gemm_f32_wmma<<<dim3(D / 64, (T * Bb) / 256), 256, 0, stream>>>(aout, W_out, b_out, out,
                                                                  T * Bb, D, D);
  (void)in_sizes; (void)n_in; (void)out_size; (void)ws_size;
}